// AM_normfree_softmax_anneal_MHE_ce_head_50568944943169
// MI455X (gfx1250) — compile-verified
//
#include <hip/hip_runtime.h>
#include <math.h>

typedef float v2f __attribute__((ext_vector_type(2)));
typedef float v8f __attribute__((ext_vector_type(8)));

#define BATCH 512
#define DIM   512
#define NCLS  50000
#define SSCALE 30.0f
#define MARGIN 0.2f
#define EPSV  1e-12f

// workspace layout (float offsets)
#define WS_RNORM 0                       // NCLS floats (padded to 50176)
#define WS_A     50176                   // 1024*512 floats
#define WS_Z     (50176 + 1024*512)      // 1024*50000 floats (cos rows 0..511, g rows 512..1023)
#define WS_ACC   (WS_Z + 1024*50000)     // 2 floats: [0]=ce sum, [1]=inter sum

// ---------------------------------------------------------------- K0: zero accumulators
__global__ void k_zero_acc(float* acc) {
    if (threadIdx.x < 2) acc[threadIdx.x] = 0.0f;
}

// ---------------------------------------------------------------- K1: rnorm[j] = 1/||W_j||
__global__ __launch_bounds__(256) void k_rnorm(const float* __restrict__ W,
                                               float* __restrict__ rnorm) {
    const int row = blockIdx.x;
    const int t   = threadIdx.x;
    const float* wr = W + (size_t)row * DIM;
    float a = wr[t], b = wr[t + 256];
    float s = a * a + b * b;
    __shared__ float red[256];
    red[t] = s; __syncthreads();
    for (int k = 128; k > 0; k >>= 1) {
        if (t < k) red[t] += red[t + k];
        __syncthreads();
    }
    if (t == 0) rnorm[row] = 1.0f / sqrtf(red[0]);
}

// ---------------------------------------------------------------- K2: build A = [emb ; rnorm[y]*W[y]]
__global__ __launch_bounds__(256) void k_build_a(const float* __restrict__ emb,
                                                 const float* __restrict__ W,
                                                 const int*   __restrict__ y,
                                                 const float* __restrict__ rnorm,
                                                 float* __restrict__ A) {
    const int row = blockIdx.x;       // 0..1023
    const int t   = threadIdx.x;      // 0..255, two elems each
    float* ar = A + (size_t)row * DIM;
    if (row < BATCH) {
        const float* er = emb + (size_t)row * DIM;
        ar[t]       = er[t];
        ar[t + 256] = er[t + 256];
    } else {
        const int i  = row - BATCH;
        const int c  = y[i];
        const float sc = rnorm[c];
        const float* wr = W + (size_t)c * DIM;
        ar[t]       = wr[t]       * sc;
        ar[t + 256] = wr[t + 256] * sc;
    }
}

// ---------------------------------------------------------------- K3: Z = (A @ W^T) * rnorm[col]
// grid = 3125 (16-col strips), block = 256 (8 waves); wave w owns rows [128w,128w+128)
__global__ __launch_bounds__(256) void k_gemm_wmma(const float* __restrict__ W,
                                                   const float* __restrict__ A,
                                                   const float* __restrict__ rnorm,
                                                   float* __restrict__ Z) {
    const int nbase = blockIdx.x * 16;
    const int lane  = threadIdx.x & 31;
    const int wave  = threadIdx.x >> 5;
    const int ln    = lane & 15;
    const int kh    = (lane >> 4) * 2;      // K sub-offset: 0 or 2
    const int col   = nbase + ln;
    const int mbase = wave * 128;

    const float* wrow = W + (size_t)col * DIM + kh;                 // B-fragment source
    const float* arow = A + ((size_t)(mbase + ln)) * DIM + kh;      // A-fragment source (tile 0)

    v8f acc[8];
    #pragma unroll
    for (int t = 0; t < 8; ++t) acc[t] = (v8f){0.f,0.f,0.f,0.f,0.f,0.f,0.f,0.f};

    for (int ks = 0; ks < DIM / 4; ++ks) {
        const int k = ks * 4;
        v2f b = *(const v2f*)(wrow + k);
        #pragma unroll
        for (int t = 0; t < 8; ++t) {
            v2f a = *(const v2f*)(arow + (size_t)t * (16 * DIM) + k);
            acc[t] = __builtin_amdgcn_wmma_f32_16x16x4_f32(
                false, a, false, b, (short)0, acc[t], false, false);
        }
    }

    const float rn = rnorm[col];
    const int rowoff = (lane >> 4) * 8;     // C/D layout: lanes 16-31 hold M=8..15
    #pragma unroll
    for (int t = 0; t < 8; ++t) {
        const int m = mbase + t * 16 + rowoff;
        float* zp = Z + (size_t)m * NCLS + col;
        #pragma unroll
        for (int v = 0; v < 8; ++v)
            zp[(size_t)v * NCLS] = acc[t][v] * rn;
    }
}

// ---------------------------------------------------------------- K4: per-row CE + MHE partials
__global__ __launch_bounds__(256) void k_reduce(const float* __restrict__ Z,
                                                const int*   __restrict__ y,
                                                float* __restrict__ acc) {
    const int i  = blockIdx.x;      // batch row
    const int t  = threadIdx.x;
    const int yi = y[i];
    const float* cosr = Z + (size_t)i * NCLS;
    const float* gr   = Z + (size_t)(BATCH + i) * NCLS;
    __shared__ float red[256];

    // pass 1: row max of logits  (S*cos - S*M at own class)
    float mx = -1e30f;
    for (int j = t; j < NCLS; j += 256) {
        float lg = SSCALE * cosr[j] - ((j == yi) ? (SSCALE * MARGIN) : 0.0f);
        mx = fmaxf(mx, lg);
    }
    red[t] = mx; __syncthreads();
    for (int k = 128; k > 0; k >>= 1) {
        if (t < k) red[t] = fmaxf(red[t], red[t + k]);
        __syncthreads();
    }
    const float rowmax = red[0]; __syncthreads();

    // pass 2: sumexp + inter-class 1/d^2
    float se = 0.0f, ip = 0.0f;
    for (int j = t; j < NCLS; j += 256) {
        float lg = SSCALE * cosr[j] - ((j == yi) ? (SSCALE * MARGIN) : 0.0f);
        se += expf(lg - rowmax);
        if (j != yi) {
            float d2 = fmaxf(2.0f - 2.0f * gr[j], EPSV);
            ip += 1.0f / d2;
        }
    }
    red[t] = se; __syncthreads();
    for (int k = 128; k > 0; k >>= 1) {
        if (t < k) red[t] += red[t + k];
        __syncthreads();
    }
    const float sumexp = red[0]; __syncthreads();

    red[t] = ip; __syncthreads();
    for (int k = 128; k > 0; k >>= 1) {
        if (t < k) red[t] += red[t + k];
        __syncthreads();
    }

    if (t == 0) {
        const float logit_y = SSCALE * (cosr[yi] - MARGIN);
        const float ce = rowmax + logf(sumexp) - logit_y;
        atomicAdd(&acc[0], ce);
        atomicAdd(&acc[1], red[0]);
    }
}

// ---------------------------------------------------------------- K5: finalize scalar
__global__ void k_finalize(const float* __restrict__ acc, float* __restrict__ out) {
    if (threadIdx.x == 0) {
        float ce    = acc[0] / (float)BATCH;
        float inter = acc[1] / ((float)BATCH * (float)(NCLS - 1));
        out[0] = ce + 1.0f * inter;   // LMD = 1.0
    }
}

extern "C" void kernel_launch(void* const* d_in, const int* in_sizes, int n_in,
                              void* d_out, int out_size, void* d_ws, size_t ws_size,
                              hipStream_t stream) {
    (void)in_sizes; (void)n_in; (void)out_size; (void)ws_size;
    const float* emb = (const float*)d_in[0];   // (512, 512)
    const float* W   = (const float*)d_in[1];   // (50000, 512)
    const int*   y   = (const int*)d_in[2];     // (512,)
    float* out  = (float*)d_out;

    float* ws_f   = (float*)d_ws;
    float* rnorm  = ws_f + WS_RNORM;
    float* A      = ws_f + WS_A;
    float* Z      = ws_f + WS_Z;
    float* acc    = ws_f + WS_ACC;

    k_zero_acc<<<1, 32, 0, stream>>>(acc);
    k_rnorm<<<NCLS, 256, 0, stream>>>(W, rnorm);
    k_build_a<<<1024, 256, 0, stream>>>(emb, W, y, rnorm, A);
    k_gemm_wmma<<<NCLS / 16, 256, 0, stream>>>(W, A, rnorm, Z);
    k_reduce<<<BATCH, 256, 0, stream>>>(Z, y, acc);
    k_finalize<<<1, 32, 0, stream>>>(acc, out);
}